// BFMatcher_12068858101989
// MI455X (gfx1250) — compile-verified
//
#include <hip/hip_runtime.h>
#include <hip/hip_bf16.h>
#include <math.h>

typedef __attribute__((ext_vector_type(2))) float v2f;
typedef __attribute__((ext_vector_type(8))) float v8f;
typedef __attribute__((ext_vector_type(4))) unsigned int u32x4;
typedef __attribute__((ext_vector_type(8))) int i32x8;
typedef __attribute__((ext_vector_type(4))) int i32x4;

#define RATIO_TEST 0.85f
#define EPS_F 1e-8f

// ---------------------------------------------------------------------------
// TDM: DMA one 16x128 f32 tile (contiguous rows, row stride 128) into LDS.
// D# built per CDNA5 ISA (group0 128b, group1 256b).
// amdgpu-toolchain builtin arity: 6 args (g0, g1, g2, g3, g_extra, cpol).
// ---------------------------------------------------------------------------
__device__ __forceinline__ void tdm_load_tile16x128(unsigned lds_off, const float* gptr) {
    unsigned long long ga = (unsigned long long)(size_t)gptr;
    u32x4 g0;
    g0[0] = 1u;                                           // count=1 (valid user D#)
    g0[1] = lds_off;                                      // lds_addr [63:32]
    g0[2] = (unsigned)(ga & 0xFFFFFFFFull);               // global_addr[31:0]
    g0[3] = (unsigned)((ga >> 32) & 0x01FFFFFFull)        // global_addr[56:32]
          | (2u << 30);                                   // type=2 ("image")
    i32x8 g1;
    g1[0] = (int)(2u << 16);        // workgroup_mask=0 | data_size=2 (4 bytes)
    g1[1] = (int)(128u << 16);      // tensor_dim0[15:0]=128 at bits[63:48]
    g1[2] = (int)(4096u << 16);     // tensor_dim0 hi=0 | tensor_dim1[15:0]=4096
    g1[3] = (int)(128u << 16);      // tensor_dim1 hi=0 | tile_dim0=128
    g1[4] = 16;                     // tile_dim1=16 | tile_dim2=0
    g1[5] = 128;                    // tensor_dim0_stride (low 32) = 128 elems
    g1[6] = 0;                      // stride0 hi | stride1 low
    g1[7] = 0;                      // stride1 hi
    i32x4 z4 = {0, 0, 0, 0};        // groups 2/3 unused (2D tensor)
    i32x8 z8 = {0, 0, 0, 0, 0, 0, 0, 0};
    __builtin_amdgcn_tensor_load_to_lds(g0, g1, z4, z4, z8, 0);
}

// ---------------------------------------------------------------------------
// Phase 1: fused GEMM (f32 WMMA) + per-row top-2 + ratio test.
//   grid = B * (N1/64), 128 threads (4 waves). Each wave owns a 16-row M-tile.
//   desc2 tiles are DMA'd into a double-buffered LDS tile by the TDM (wave 0),
//   overlapped with the WMMA chain of the previous tile.
// ---------------------------------------------------------------------------
__global__ __launch_bounds__(128)
void bf_match_top2(const float* __restrict__ d1, const float* __restrict__ d2,
                   int* __restrict__ mask_ws, int* __restrict__ idx_ws) {
    constexpr int N1 = 4096, N2 = 4096, D = 128;
    constexpr int T = N2 / 16;                      // 256 tiles
    const int wgPerBatch = N1 / 64;                 // 64
    const int b    = blockIdx.x / wgPerBatch;
    const int mt   = blockIdx.x % wgPerBatch;
    const int wave = threadIdx.x >> 5;
    const int lane = threadIdx.x & 31;
    const int rowBase = mt * 64 + wave * 16;
    const int m  = lane & 15;                       // A row / B,C column
    const int kh = lane >> 4;                       // K-pair select: 0->{0,1}, 1->{2,3}

    // Full K=128 A-panel for this wave's 16 rows: 32 chunks x v2f = 64 VGPRs.
    const float* a_ptr = d1 + ((size_t)b * N1 + rowBase + m) * D + kh * 2;
    v2f a[32];
#pragma unroll
    for (int c = 0; c < 32; ++c)
        a[c] = *(const v2f*)(a_ptr + c * 4);

    __shared__ __align__(16) float sB[2][16 * D];   // 2 x 8KB double buffer

    float t1[8], t2[8];
    int   i1[8];
#pragma unroll
    for (int s = 0; s < 8; ++s) { t1[s] = -INFINITY; t2[s] = -INFINITY; i1[s] = 0; }

    const float* d2b = d2 + (size_t)b * N2 * D;

    // Prime the pipeline: TDM tile 0 -> buffer 0 (one wave issues; EXEC ignored).
    if (wave == 0)
        tdm_load_tile16x128((unsigned)(size_t)(void*)&sB[0][0], d2b);

    for (int t = 0; t < T; ++t) {
        if (wave == 0)
            __builtin_amdgcn_s_wait_tensorcnt(0);   // tile t landed in LDS
        __syncthreads();                            // publish tile t; prev compute done

        // Prefetch tile t+1 into the other buffer while we compute tile t.
        if (wave == 0 && (t + 1) < T)
            tdm_load_tile16x128((unsigned)(size_t)(void*)&sB[(t + 1) & 1][0],
                                d2b + (size_t)(t + 1) * 16 * D);

        // Preload all 32 B fragments (B[k][n] = desc2[n][k]) into registers:
        // one LDS clause, single wait, then an uninterrupted WMMA chain.
        const float* bbase = &sB[t & 1][0] + (size_t)m * D + kh * 2;
        v2f bfr[32];
#pragma unroll
        for (int c = 0; c < 32; ++c)
            bfr[c] = *(const v2f*)(bbase + c * 4);

        // Dual accumulators break the RAW accumulation chain (2x XDL ILP).
        v8f acc0 = {}, acc1 = {};
#pragma unroll
        for (int c = 0; c < 32; c += 2) {
            acc0 = __builtin_amdgcn_wmma_f32_16x16x4_f32(
                false, a[c],     false, bfr[c],     (short)0, acc0, false, false);
            acc1 = __builtin_amdgcn_wmma_f32_16x16x4_f32(
                false, a[c + 1], false, bfr[c + 1], (short)0, acc1, false, false);
        }

        // Fused top-2 update: each lane covers columns n ≡ (lane&15) (mod 16).
        const int col = t * 16 + m;
#pragma unroll
        for (int s = 0; s < 8; ++s) {
            float v = acc0[s] + acc1[s];
            bool gt1 = v > t1[s];
            float nv2 = gt1 ? t1[s] : ((v > t2[s]) ? v : t2[s]);
            if (gt1) { t1[s] = v; i1[s] = col; }
            t2[s] = nv2;
        }
    }

    // Cross-lane top-2 merge over the 16-lane half (columns mod 16).
    // C layout: slot s, lanes 0-15 -> row rowBase+s ; lanes 16-31 -> rowBase+8+s.
#pragma unroll
    for (int s = 0; s < 8; ++s) {
        float a1 = t1[s], a2 = t2[s];
        int   ai = i1[s];
#pragma unroll
        for (int off = 1; off < 16; off <<= 1) {
            float b1 = __shfl_xor(a1, off, 32);
            float b2 = __shfl_xor(a2, off, 32);
            int   bi = __shfl_xor(ai, off, 32);
            bool aw = (a1 > b1) || (a1 == b1 && ai < bi);
            float n1 = aw ? a1 : b1;
            int   ni = aw ? ai : bi;
            float n2 = aw ? fmaxf(a2, b1) : fmaxf(b2, a1);
            a1 = n1; ai = ni; a2 = n2;
        }
        if (m == 0) {                               // lanes 0 and 16 write
            int row = rowBase + (kh ? 8 : 0) + s;
            float ratio = a1 / (a2 + EPS_F);
            mask_ws[(size_t)b * N1 + row] = (ratio < RATIO_TEST) ? 1 : 0;
            idx_ws [(size_t)b * N1 + row] = ai;
        }
    }
}

// ---------------------------------------------------------------------------
// Phase 2: per-batch stable compaction (zero-fill, Hillis-Steele scan, scatter).
// ---------------------------------------------------------------------------
__global__ __launch_bounds__(1024)
void bf_match_compact(const int* __restrict__ mask_ws, const int* __restrict__ idx_ws,
                      int* __restrict__ out) {
    constexpr int N1 = 4096;
    const int b = blockIdx.x;
    const int t = threadIdx.x;
    __shared__ int ssum[1024];

    int* outb = out + (size_t)b * N1 * 2;
    for (int i = t; i < N1 * 2; i += 1024) outb[i] = 0;

    const int* mb = mask_ws + (size_t)b * N1;
    const int* ib = idx_ws  + (size_t)b * N1;

    int mrow[4];
    int s = 0;
#pragma unroll
    for (int j = 0; j < 4; ++j) { mrow[j] = mb[t * 4 + j]; s += mrow[j]; }

    ssum[t] = s;
    __syncthreads();
    int acc = s;
    for (int off = 1; off < 1024; off <<= 1) {
        int v = (t >= off) ? ssum[t - off] : 0;
        __syncthreads();
        acc += v;
        ssum[t] = acc;
        __syncthreads();
    }
    int pos = acc - s;                              // stable exclusive prefix
#pragma unroll
    for (int j = 0; j < 4; ++j) {
        int row = t * 4 + j;
        if (mrow[j]) {
            outb[pos * 2]     = row;
            outb[pos * 2 + 1] = ib[row];
            ++pos;
        }
    }
}

extern "C" void kernel_launch(void* const* d_in, const int* in_sizes, int n_in,
                              void* d_out, int out_size, void* d_ws, size_t ws_size,
                              hipStream_t stream) {
    constexpr int B = 4, N1 = 4096;
    const float* d1 = (const float*)d_in[0];
    const float* d2 = (const float*)d_in[1];
    int* out = (int*)d_out;                         // int32 matches, reinterpreted
    int* mask_ws = (int*)d_ws;                      // B*N1 ints
    int* idx_ws  = mask_ws + (size_t)B * N1;        // B*N1 ints  (128 KB total)

    bf_match_top2<<<B * (N1 / 64), 128, 0, stream>>>(d1, d2, mask_ws, idx_ws);
    bf_match_compact<<<B, 1024, 0, stream>>>(mask_ws, idx_ws, out);
}